// SampledGraphConvolution_28767690949397
// MI455X (gfx1250) — compile-verified
//
#include <hip/hip_runtime.h>
#include <math.h>

// Problem constants (match reference)
#define NN   16384
#define DIN  64
#define DOUT 32
#define EPSV 1e-12f

typedef float v2f __attribute__((ext_vector_type(2)));
typedef float v4f __attribute__((ext_vector_type(4)));
typedef float v8f __attribute__((ext_vector_type(8)));
typedef unsigned int v4u __attribute__((ext_vector_type(4)));
typedef int v4i __attribute__((ext_vector_type(4)));
typedef int v8i __attribute__((ext_vector_type(8)));

// ---------------------------------------------------------------------------
// Kernel 1: per-row r[n] = rowsum(x)/max(||x||,eps); h = X @ W (64x32 in LDS)
// ---------------------------------------------------------------------------
__global__ __launch_bounds__(256) void gcn_row(const float* __restrict__ X,
                                               const float* __restrict__ W,
                                               float* __restrict__ r,
                                               float* __restrict__ h) {
  __shared__ float lW[DIN * DOUT];
  for (int i = threadIdx.x; i < DIN * DOUT; i += 256) lW[i] = W[i];
  __syncthreads();

  const int n = blockIdx.x * 256 + threadIdx.x;
  const float* xp = X + (size_t)n * DIN;

  float ss = 0.f, s = 0.f;
#pragma unroll
  for (int d = 0; d < DIN; ++d) {
    float xd = xp[d];
    ss = fmaf(xd, xd, ss);
    s += xd;
  }
  float nrm = fmaxf(sqrtf(ss), EPSV);
  r[n] = s / nrm;

  float acc[DOUT];
#pragma unroll
  for (int l = 0; l < DOUT; ++l) acc[l] = 0.f;
  for (int d = 0; d < DIN; ++d) {   // rolled outer (x re-read hits L0)
    float xd = xp[d];
#pragma unroll
    for (int l = 0; l < DOUT; ++l) acc[l] = fmaf(xd, lW[d * DOUT + l], acc[l]);
  }
  float* hp = h + (size_t)n * DOUT;
#pragma unroll
  for (int l = 0; l < DOUT; ++l) hp[l] = acc[l];
}

// ---------------------------------------------------------------------------
// Kernel 2: one streaming pass over A. For 1024-row chunk `blockIdx.y`:
//   s2part[y][j] = sum A[n,j]^2 ,  tpart[y][j] = sum A[n,j]*r[n]
// Deterministic (no float atomics); coalesced 128B/wave per row.
// ---------------------------------------------------------------------------
__global__ __launch_bounds__(256) void gcn_col(const float* __restrict__ A,
                                               const float* __restrict__ r,
                                               float* __restrict__ s2part,
                                               float* __restrict__ tpart) {
  __shared__ float lr[1024];
  const int j  = blockIdx.x * 256 + threadIdx.x;
  const int n0 = blockIdx.y * 1024;
  for (int i = threadIdx.x; i < 1024; i += 256) lr[i] = r[n0 + i];
  __syncthreads();

  float a2 = 0.f, ar = 0.f;
  const float* p = A + (size_t)n0 * NN + j;
#pragma unroll 4
  for (int i = 0; i < 1024; ++i) {
    float a = __builtin_nontemporal_load(p + (size_t)i * NN);
    a2 = fmaf(a, a, a2);
    ar = fmaf(a, lr[i], ar);
  }
  s2part[(size_t)blockIdx.y * NN + j] = a2;
  tpart [(size_t)blockIdx.y * NN + j] = ar;
}

// ---------------------------------------------------------------------------
// Kernel 3: mix[j] = t[j]/max(sqrt(s2[j]),eps) ; S = sum_j mix[j]
// Single block, fixed reduction tree -> deterministic.
// ---------------------------------------------------------------------------
__global__ __launch_bounds__(1024) void gcn_mix(const float* __restrict__ s2part,
                                                const float* __restrict__ tpart,
                                                float* __restrict__ mix,
                                                float* __restrict__ S) {
  __shared__ float red[1024];
  float loc = 0.f;
  for (int j = threadIdx.x; j < NN; j += 1024) {
    float a2 = 0.f, ar = 0.f;
#pragma unroll
    for (int c = 0; c < 16; ++c) {
      a2 += s2part[(size_t)c * NN + j];
      ar += tpart [(size_t)c * NN + j];
    }
    float mj = ar / fmaxf(sqrtf(a2), EPSV);
    mix[j] = mj;
    loc += mj;
  }
  red[threadIdx.x] = loc;
  __syncthreads();
  for (int off = 512; off > 0; off >>= 1) {
    if (threadIdx.x < off) red[threadIdx.x] += red[threadIdx.x + off];
    __syncthreads();
  }
  if (threadIdx.x == 0) S[0] = red[0];
}

// ---------------------------------------------------------------------------
// Kernel 4: h2[n,:] = h[n,:] * mix[n] / S   (folds new_support = A*diag(m))
// ---------------------------------------------------------------------------
__global__ __launch_bounds__(256) void gcn_scale(float* __restrict__ h,
                                                 const float* __restrict__ mix,
                                                 const float* __restrict__ S) {
  const int idx = blockIdx.x * 256 + threadIdx.x;  // N*32 total
  const float inv = 1.0f / S[0];
  h[idx] *= mix[idx >> 5] * inv;
}

// ---------------------------------------------------------------------------
// Kernel 5: out = leaky_relu(A @ h2)  — 16384x16384x32 fp32 via
// V_WMMA_F32_16X16X4_F32.  128 threads = 4 waves; block tiles 64 rows x K=64
// chunks, double-buffered LDS; each wave owns a 16x32 output tile.
// A tile (the 1 GB streaming operand) is moved by the Tensor Data Mover:
// wave 0 issues tensor_load_to_lds with a D# that pads each 64-DWORD row by
// 4 DWORDs (=> LDA 68, bank-conflict-free ds_load_b64 fragments), tracked on
// TENSORcnt and overlapped with the WMMA stream. B tile (8 KB) keeps the
// register path since it is stored transposed.
// ---------------------------------------------------------------------------
#define BK  64
#define BM  64
#define LDA 68   // floats; bank = (4r+k)%64 -> conflict-free 16x4 A fragments
#define LDB 68

__device__ __forceinline__ void tdm_load_tile_a(const float* gptr, unsigned lds_byte) {
  // D# group 0: count=1 (valid user descriptor), lds_addr, 57-bit global addr, type=2
  unsigned long long ga = (unsigned long long)(uintptr_t)gptr;
  v4u g0;
  g0.x = 1u;                                   // count=1
  g0.y = lds_byte;                             // lds_addr (bytes)
  g0.z = (unsigned)ga;                         // global_addr[31:0]
  g0.w = (unsigned)((ga >> 32) & 0x01FFFFFFu)  // global_addr[56:32]
         | (2u << 30);                         // type=2 ("image")
  // D# group 1: mask=0, data_size=2 (4B), pad_enable=1,
  // pad_interval=5 (every 64 DWORDs), pad_amount=3 (4 DWORDs) -> row stride 68
  v8i g1;
  g1[0] = (int)((2u << 16) | (1u << 20) | (5u << 22) | (3u << 25));
  g1[1] = (int)((unsigned)(NN & 0xFFFF) << 16);        // tensor_dim0[15:0]
  g1[2] = (int)(((unsigned)NN >> 16) |                 // tensor_dim0[31:16]
                ((unsigned)(NN & 0xFFFF) << 16));      // tensor_dim1[15:0]
  g1[3] = (int)(((unsigned)NN >> 16) | (64u << 16));   // tensor_dim1[31:16], tile_dim0=64
  g1[4] = 64;                                          // tile_dim1=64, tile_dim2=0
  g1[5] = NN;                                          // tensor_dim0_stride[31:0]
  g1[6] = 0;                                           // stride0 hi, stride1 lo
  g1[7] = 0;                                           // stride1 hi
  v4i gz4 = {0, 0, 0, 0};                              // groups 2/3: 2-D tile, unused
  v8i gz8 = {0, 0, 0, 0, 0, 0, 0, 0};                  // extended words: unused
  __builtin_amdgcn_tensor_load_to_lds(g0, g1, gz4, gz4, gz8, 0);
}

__global__ __launch_bounds__(128) void gcn_gemm(const float* __restrict__ A,
                                                const float* __restrict__ H,
                                                float* __restrict__ out) {
  __shared__ float sA[2][BM * LDA];   // 2*64*68*4  = 34816 B (TDM-filled)
  __shared__ float sB[2][DOUT * LDB]; // 2*32*68*4  = 17408 B (transposed [col][k])

  const int tid  = threadIdx.x;
  const int wave = tid >> 5;
  const int lane = tid & 31;
  const int rowb = blockIdx.x * BM;

  v8f acc0 = {0.f,0.f,0.f,0.f,0.f,0.f,0.f,0.f};
  v8f acc1 = {0.f,0.f,0.f,0.f,0.f,0.f,0.f,0.f};

  v4f rb[4];   // B tile staging: 64x32 floats / 128 threads = 16 floats each

  const unsigned ldsA0 = (unsigned)(uintptr_t)&sA[0][0];
  const unsigned ldsA1 = (unsigned)(uintptr_t)&sA[1][0];

  auto loadB = [&](int kc) {
    const float* hp = H + (size_t)kc * DOUT;
    int i = tid >> 1, j0 = (tid & 1) * 16;   // k-row 0..63, col half
#pragma unroll
    for (int q = 0; q < 4; ++q)
      rb[q] = *(const v4f*)(hp + i * DOUT + j0 + q * 4);
  };
  auto storeB = [&](int buf) {
    int i = tid >> 1, j0 = (tid & 1) * 16;
#pragma unroll
    for (int q = 0; q < 4; ++q)
#pragma unroll
      for (int e = 0; e < 4; ++e)
        sB[buf][(j0 + q * 4 + e) * LDB + i] = rb[q][e];  // transpose into [col][k]
  };

  // Prologue: chunk 0 in flight (TDM for A, registers for B)
  loadB(0);
  if (tid < 32) tdm_load_tile_a(A + (size_t)rowb * NN, ldsA0);
  storeB(0);
  if (tid < 32) __builtin_amdgcn_s_wait_tensorcnt(0);
  __syncthreads();

  const int koff = (lane >> 4) << 1;            // lanes 0-15 -> K+0/1, 16-31 -> K+2/3
  const int arow = (wave << 4) + (lane & 15);   // wave's 16-row tile

  const int NCHUNK = NN / BK;                   // 256
  for (int c = 0; c < NCHUNK; ++c) {
    const int cur = c & 1;
    if (c + 1 < NCHUNK) {                       // next chunk in flight over the WMMAs
      loadB((c + 1) * BK);
      if (tid < 32)
        tdm_load_tile_a(A + (size_t)rowb * NN + (size_t)(c + 1) * BK,
                        ((c + 1) & 1) ? ldsA1 : ldsA0);
    }

    const float* pa  = &sA[cur][arow * LDA + koff];
    const float* pb0 = &sB[cur][(lane & 15) * LDB + koff];
    const float* pb1 = &sB[cur][((lane & 15) + 16) * LDB + koff];
#pragma unroll
    for (int s = 0; s < BK / 4; ++s) {
      v2f af = *(const v2f*)(pa  + s * 4);   // A frag 16x4  (ds_load_b64)
      v2f b0 = *(const v2f*)(pb0 + s * 4);   // B frag 4x16, cols 0-15
      v2f b1 = *(const v2f*)(pb1 + s * 4);   // B frag 4x16, cols 16-31
      acc0 = __builtin_amdgcn_wmma_f32_16x16x4_f32(false, af, false, b0,
                                                   (short)0, acc0, false, false);
      acc1 = __builtin_amdgcn_wmma_f32_16x16x4_f32(false, af, false, b1,
                                                   (short)0, acc1, false, false);
    }
    if (c + 1 < NCHUNK) storeB((c + 1) & 1);  // buf last read at iter c-1 (pre-barrier)
    if (tid < 32) __builtin_amdgcn_s_wait_tensorcnt(0);
    __syncthreads();
  }

  // C/D layout: VGPR v holds M=v (lanes 0-15) / M=v+8 (lanes 16-31), N=lane%16
  const int rbase = rowb + (wave << 4) + ((lane >> 4) << 3);
  const int col   = lane & 15;
#pragma unroll
  for (int v = 0; v < 8; ++v) {
    float x0 = acc0[v];
    float x1 = acc1[v];
    x0 = (x0 >= 0.f) ? x0 : 0.01f * x0;   // LeakyReLU
    x1 = (x1 >= 0.f) ? x1 : 0.01f * x1;
    size_t ro = (size_t)(rbase + v) * DOUT;
    out[ro + col]      = x0;
    out[ro + col + 16] = x1;
  }
}

// ---------------------------------------------------------------------------
extern "C" void kernel_launch(void* const* d_in, const int* in_sizes, int n_in,
                              void* d_out, int out_size, void* d_ws, size_t ws_size,
                              hipStream_t stream) {
  (void)in_sizes; (void)n_in; (void)out_size; (void)ws_size;
  const float* A = (const float*)d_in[0];   // [N,N]
  const float* X = (const float*)d_in[1];   // [N,64]
  const float* W = (const float*)d_in[2];   // [64,32]
  float* out = (float*)d_out;               // [N,32]

  float* ws     = (float*)d_ws;             // ~4.3 MB used
  float* r      = ws;                       // N
  float* s2part = r + NN;                   // 16*N
  float* tpart  = s2part + 16 * NN;         // 16*N
  float* mix    = tpart + 16 * NN;          // N
  float* S      = mix + NN;                 // 1 (+pad to 16 for alignment)
  float* h      = S + 16;                   // N*32  (scaled in place -> h2)

  gcn_row  <<<NN / 256, 256, 0, stream>>>(X, W, r, h);
  gcn_col  <<<dim3(NN / 256, 16), 256, 0, stream>>>(A, r, s2part, tpart);
  gcn_mix  <<<1, 1024, 0, stream>>>(s2part, tpart, mix, S);
  gcn_scale<<<NN * DOUT / 256, 256, 0, stream>>>(h, mix, S);
  gcn_gemm <<<NN / BM, 128, 0, stream>>>(A, h, out);
}